// ResHAN_15882789060984
// MI455X (gfx1250) — compile-verified
//
#include <hip/hip_runtime.h>
#include <math.h>

// ResHAN on gfx1250: fp32 WMMA (16x16x4) for all GEMM stages, f32 global
// atomics for segment softmax / scatter. Memory-bound workload -> keep fp32.

#define HID 128
#define NH 8
#define DH 16
#define OUTC 64
#define NSL 0.2f

typedef float v2f __attribute__((ext_vector_type(2)));
typedef float v8f __attribute__((ext_vector_type(8)));

__device__ __forceinline__ v8f wmma4(v2f a, v2f b, v8f c) {
  // D = A(16x4,f32) * B(4x16,f32) + C(16x16,f32)
  return __builtin_amdgcn_wmma_f32_16x16x4_f32(false, a, false, b,
                                               (short)0, c, false, false);
}

__device__ __forceinline__ void atomicMaxF(float* addr, float val) {
  if (val >= 0.0f) atomicMax((int*)addr, __float_as_int(val));
  else             atomicMin((unsigned int*)addr, __float_as_uint(val));
}

// ---------------- GEMM: Y[M,128] = A[M,128] @ W[128,128] + bias ----------------
__global__ __launch_bounds__(128) void gemm128_bias(
    const float* __restrict__ A, const float* __restrict__ W,
    const float* __restrict__ bias, float* __restrict__ Y, int M) {
  const int lane = threadIdx.x & 31;
  const int rb = blockIdx.x * 4 + (threadIdx.x >> 5);
  if (rb * 16 >= M) return;                 // wave-uniform: EXEC stays full
  const int m0 = rb * 16;
  const int half = lane >> 4;
  const int lr = lane & 15;
  v8f acc[8];
#pragma unroll
  for (int t = 0; t < 8; ++t) acc[t] = v8f{0,0,0,0,0,0,0,0};
  const float* arow = A + (size_t)(m0 + lr) * HID;
  for (int k0 = 0; k0 < HID; k0 += 4) {
    v2f a;
    a.x = arow[k0 + 2 * half + 0];
    a.y = arow[k0 + 2 * half + 1];
    const float* wr = W + (size_t)(k0 + 2 * half) * HID + lr;
#pragma unroll
    for (int t = 0; t < 8; ++t) {
      v2f b;
      b.x = wr[t * 16];
      b.y = wr[HID + t * 16];
      acc[t] = wmma4(a, b, acc[t]);
    }
  }
#pragma unroll
  for (int t = 0; t < 8; ++t)
#pragma unroll
    for (int r = 0; r < 8; ++r) {
      const int row = m0 + r + 8 * half;
      const int col = t * 16 + lr;
      Y[(size_t)row * HID + col] = acc[t][r] + bias[col];
    }
}

// ------- tanh-score: score += sum_n sum_c tanh(A@Wk + bk)[n,c] * q[c] -------
__global__ __launch_bounds__(128) void gemm128_tanh_score(
    const float* __restrict__ A, const float* __restrict__ W,
    const float* __restrict__ bias, const float* __restrict__ q,
    float* __restrict__ score, int M) {
  const int lane = threadIdx.x & 31;
  const int rb = blockIdx.x * 4 + (threadIdx.x >> 5);
  if (rb * 16 >= M) return;
  const int m0 = rb * 16;
  const int half = lane >> 4;
  const int lr = lane & 15;
  v8f acc[8];
#pragma unroll
  for (int t = 0; t < 8; ++t) acc[t] = v8f{0,0,0,0,0,0,0,0};
  const float* arow = A + (size_t)(m0 + lr) * HID;
  for (int k0 = 0; k0 < HID; k0 += 4) {
    v2f a;
    a.x = arow[k0 + 2 * half + 0];
    a.y = arow[k0 + 2 * half + 1];
    const float* wr = W + (size_t)(k0 + 2 * half) * HID + lr;
#pragma unroll
    for (int t = 0; t < 8; ++t) {
      v2f b;
      b.x = wr[t * 16];
      b.y = wr[HID + t * 16];
      acc[t] = wmma4(a, b, acc[t]);
    }
  }
  float local = 0.0f;
#pragma unroll
  for (int t = 0; t < 8; ++t) {
    const int col = t * 16 + lr;
    const float bc = bias[col], qc = q[col];
#pragma unroll
    for (int r = 0; r < 8; ++r) local += tanhf(acc[t][r] + bc) * qc;
  }
#pragma unroll
  for (int off = 16; off > 0; off >>= 1) local += __shfl_xor(local, off, 32);
  if (lane == 0) atomicAdd(score, local);
}

// ----- final: OUT[M,64] = (b0*Agg + b1*Adg)[M,128] @ Wl[128,64] + bias -----
__global__ __launch_bounds__(128) void gemm_final(
    const float* __restrict__ Agg, const float* __restrict__ Adg,
    const float* __restrict__ beta, const float* __restrict__ W,
    const float* __restrict__ bias, float* __restrict__ Y, int M) {
  const int lane = threadIdx.x & 31;
  const int rb = blockIdx.x * 4 + (threadIdx.x >> 5);
  if (rb * 16 >= M) return;
  const float b0 = beta[0], b1 = beta[1];
  const int m0 = rb * 16;
  const int half = lane >> 4;
  const int lr = lane & 15;
  v8f acc[4];
#pragma unroll
  for (int t = 0; t < 4; ++t) acc[t] = v8f{0,0,0,0,0,0,0,0};
  const float* g = Agg + (size_t)(m0 + lr) * HID;
  const float* d = Adg + (size_t)(m0 + lr) * HID;
  for (int k0 = 0; k0 < HID; k0 += 4) {
    const int k = k0 + 2 * half;
    v2f a;
    a.x = b0 * g[k]     + b1 * d[k];
    a.y = b0 * g[k + 1] + b1 * d[k + 1];
    const float* wr = W + (size_t)k * OUTC + lr;
#pragma unroll
    for (int t = 0; t < 4; ++t) {
      v2f b;
      b.x = wr[t * 16];
      b.y = wr[OUTC + t * 16];
      acc[t] = wmma4(a, b, acc[t]);
    }
  }
#pragma unroll
  for (int t = 0; t < 4; ++t)
#pragma unroll
    for (int r = 0; r < 8; ++r) {
      const int row = m0 + r + 8 * half;
      const int col = t * 16 + lr;
      Y[(size_t)row * OUTC + col] = acc[t][r] + bias[col];
    }
}

// ---------------- per-node attention logits: e[n,h] = <feat[n,h,:], att[h,:]> ----------------
__global__ void node_att(const float* __restrict__ feat, const float* __restrict__ att,
                         float* __restrict__ e, int N) {
  const int i = blockIdx.x * blockDim.x + threadIdx.x;
  if (i >= N * NH) return;
  const int n = i >> 3, h = i & 7;
  const float* f = feat + (size_t)n * HID + h * DH;
  const float* a = att + h * DH;
  float s = 0.0f;
#pragma unroll
  for (int d = 0; d < DH; ++d) s += f[d] * a[d];
  e[i] = s;
}

__global__ void fill_f32(float* __restrict__ p, float v, long long n) {
  const long long i = (long long)blockIdx.x * blockDim.x + threadIdx.x;
  if (i < n) p[i] = v;
}

__global__ void edge_max(const int* __restrict__ src, const int* __restrict__ dst,
                         const float* __restrict__ es, const float* __restrict__ ed,
                         float* __restrict__ m, int E) {
  const int i = blockIdx.x * blockDim.x + threadIdx.x;
  if (i >= E * NH) return;
  const int e = i >> 3, h = i & 7;
  const int s = src[e], d = dst[e];
  float a = es[s * NH + h] + ed[d * NH + h];
  a = (a >= 0.0f) ? a : NSL * a;
  atomicMaxF(&m[d * NH + h], a);
}

__global__ void fix_m(float* __restrict__ m, long long n) {
  const long long i = (long long)blockIdx.x * blockDim.x + threadIdx.x;
  if (i < n) { float v = m[i]; m[i] = isfinite(v) ? v : 0.0f; }
}

__global__ void edge_expsum(const int* __restrict__ src, const int* __restrict__ dst,
                            const float* __restrict__ es, const float* __restrict__ ed,
                            const float* __restrict__ m, float* __restrict__ denom, int E) {
  const int i = blockIdx.x * blockDim.x + threadIdx.x;
  if (i >= E * NH) return;
  const int e = i >> 3, h = i & 7;
  const int s = src[e], d = dst[e];
  float a = es[s * NH + h] + ed[d * NH + h];
  a = (a >= 0.0f) ? a : NSL * a;
  atomicAdd(&denom[d * NH + h], expf(a - m[d * NH + h]));
}

__global__ void edge_scatter(const int* __restrict__ src, const int* __restrict__ dst,
                             const float* __restrict__ es, const float* __restrict__ ed,
                             const float* __restrict__ m, const float* __restrict__ denom,
                             const float* __restrict__ xs, float* __restrict__ out, int E) {
  const long long i = (long long)blockIdx.x * blockDim.x + threadIdx.x;
  if (i >= (long long)E * HID) return;
  const int e = (int)(i >> 7);
  const int c = (int)(i & 127);
  const int h = c >> 4;
  const int s = src[e], d = dst[e];
  float a = es[s * NH + h] + ed[d * NH + h];
  a = (a >= 0.0f) ? a : NSL * a;
  const float w = expf(a - m[d * NH + h]) / (denom[d * NH + h] + 1e-16f);
  atomicAdd(&out[(size_t)d * HID + c], xs[(size_t)s * HID + c] * w);
}

__global__ void relu_inplace(float* __restrict__ p, long long n) {
  const long long i = (long long)blockIdx.x * blockDim.x + threadIdx.x;
  if (i < n) p[i] = fmaxf(p[i], 0.0f);
}

__global__ void beta_kernel(const float* __restrict__ score, float* __restrict__ beta,
                            float invN) {
  const float s0 = score[0] * invN, s1 = score[1] * invN;
  const float mx = fmaxf(s0, s1);
  const float e0 = expf(s0 - mx), e1 = expf(s1 - mx);
  const float inv = 1.0f / (e0 + e1);
  beta[0] = e0 * inv;
  beta[1] = e1 * inv;
}

extern "C" void kernel_launch(void* const* d_in, const int* in_sizes, int n_in,
                              void* d_out, int out_size, void* d_ws, size_t ws_size,
                              hipStream_t stream) {
  const float* x_gene = (const float*)d_in[0];
  const float* x_dis  = (const float*)d_in[1];
  const int* eg_src   = (const int*)d_in[2];
  const int* eg_dst   = (const int*)d_in[3];
  const int* edg_src  = (const int*)d_in[4];
  const int* edg_dst  = (const int*)d_in[5];
  const float* W_gene = (const float*)d_in[6];
  const float* b_gene = (const float*)d_in[7];
  const float* W_dis  = (const float*)d_in[8];
  const float* b_dis  = (const float*)d_in[9];
  const float* a_s_gg = (const float*)d_in[10];
  const float* a_d_gg = (const float*)d_in[11];
  const float* a_s_dg = (const float*)d_in[12];
  const float* a_d_dg = (const float*)d_in[13];
  const float* Wk     = (const float*)d_in[14];
  const float* bk     = (const float*)d_in[15];
  const float* q      = (const float*)d_in[16];
  const float* W_lin  = (const float*)d_in[17];
  const float* b_lin  = (const float*)d_in[18];
  float* out = (float*)d_out;

  const int NG  = in_sizes[0] / HID;
  const int ND  = in_sizes[1] / HID;
  const int EG  = in_sizes[2];
  const int EDG = in_sizes[4];

  // workspace layout (floats)
  float* ws = (float*)d_ws;
  size_t off = 0;
  float* hg    = ws + off; off += (size_t)NG * HID;
  float* hd    = ws + off; off += (size_t)ND * HID;
  float* esgg  = ws + off; off += (size_t)NG * NH;
  float* edgg  = ws + off; off += (size_t)NG * NH;
  float* esdg  = ws + off; off += (size_t)ND * NH;
  float* eddg  = ws + off; off += (size_t)NG * NH;
  float* mgg   = ws + off; off += (size_t)NG * NH;
  float* dngg  = ws + off; off += (size_t)NG * NH;
  float* mdg   = ws + off; off += (size_t)NG * NH;
  float* dndg  = ws + off; off += (size_t)NG * NH;
  float* ogg   = ws + off; off += (size_t)NG * HID;
  float* odg   = ws + off; off += (size_t)NG * HID;
  float* score = ws + off; off += 8;
  float* beta  = ws + off; off += 8;

  const int T = 256;
  #define CDIV(a, b) (((a) + (b) - 1) / (b))
  const int gemmBlkG = CDIV(NG / 16, 4);
  const int gemmBlkD = CDIV(ND / 16, 4);

  // 1) projections (WMMA fp32)
  gemm128_bias<<<gemmBlkG, 128, 0, stream>>>(x_gene, W_gene, b_gene, hg, NG);
  gemm128_bias<<<gemmBlkD, 128, 0, stream>>>(x_dis,  W_dis,  b_dis,  hd, ND);

  // 2) per-node attention logits
  node_att<<<CDIV(NG * NH, T), T, 0, stream>>>(hg, a_s_gg, esgg, NG);
  node_att<<<CDIV(NG * NH, T), T, 0, stream>>>(hg, a_d_gg, edgg, NG);
  node_att<<<CDIV(ND * NH, T), T, 0, stream>>>(hd, a_s_dg, esdg, ND);
  node_att<<<CDIV(NG * NH, T), T, 0, stream>>>(hg, a_d_dg, eddg, NG);

  // 3) init accumulators (ws is poisoned by the harness)
  const long long ngh = (long long)NG * NH;
  fill_f32<<<CDIV((int)ngh, T), T, 0, stream>>>(mgg, -INFINITY, ngh);
  fill_f32<<<CDIV((int)ngh, T), T, 0, stream>>>(mdg, -INFINITY, ngh);
  fill_f32<<<CDIV((int)ngh, T), T, 0, stream>>>(dngg, 0.0f, ngh);
  fill_f32<<<CDIV((int)ngh, T), T, 0, stream>>>(dndg, 0.0f, ngh);
  const long long ngc = (long long)NG * HID;
  fill_f32<<<(unsigned)CDIV(ngc, (long long)T), T, 0, stream>>>(ogg, 0.0f, ngc);
  fill_f32<<<(unsigned)CDIV(ngc, (long long)T), T, 0, stream>>>(odg, 0.0f, ngc);
  fill_f32<<<1, 32, 0, stream>>>(score, 0.0f, 2);

  // 4) segment softmax pass 1: max
  edge_max<<<CDIV(EG * NH, T), T, 0, stream>>>(eg_src, eg_dst, esgg, edgg, mgg, EG);
  edge_max<<<CDIV(EDG * NH, T), T, 0, stream>>>(edg_src, edg_dst, esdg, eddg, mdg, EDG);
  fix_m<<<CDIV((int)ngh, T), T, 0, stream>>>(mgg, ngh);
  fix_m<<<CDIV((int)ngh, T), T, 0, stream>>>(mdg, ngh);

  // 5) pass 2: sum of exp
  edge_expsum<<<CDIV(EG * NH, T), T, 0, stream>>>(eg_src, eg_dst, esgg, edgg, mgg, dngg, EG);
  edge_expsum<<<CDIV(EDG * NH, T), T, 0, stream>>>(edg_src, edg_dst, esdg, eddg, mdg, dndg, EDG);

  // 6) pass 3: weighted message scatter (recompute exp; avoids 51MB of per-edge state)
  const long long egw  = (long long)EG * HID;
  const long long edgw = (long long)EDG * HID;
  edge_scatter<<<(unsigned)CDIV(egw, (long long)T), T, 0, stream>>>(
      eg_src, eg_dst, esgg, edgg, mgg, dngg, hg, ogg, EG);
  edge_scatter<<<(unsigned)CDIV(edgw, (long long)T), T, 0, stream>>>(
      edg_src, edg_dst, esdg, eddg, mdg, dndg, hd, odg, EDG);

  // 7) ReLU
  relu_inplace<<<(unsigned)CDIV(ngc, (long long)T), T, 0, stream>>>(ogg, ngc);
  relu_inplace<<<(unsigned)CDIV(ngc, (long long)T), T, 0, stream>>>(odg, ngc);

  // 8) semantic attention scores (WMMA + tanh + wave reduce)
  gemm128_tanh_score<<<gemmBlkG, 128, 0, stream>>>(ogg, Wk, bk, q, score + 0, NG);
  gemm128_tanh_score<<<gemmBlkG, 128, 0, stream>>>(odg, Wk, bk, q, score + 1, NG);
  beta_kernel<<<1, 1, 0, stream>>>(score, beta, 1.0f / (float)NG);

  // 9) beta-combined final projection (WMMA)
  gemm_final<<<gemmBlkG, 128, 0, stream>>>(ogg, odg, beta, W_lin, b_lin, out, NG);
  (void)n_in; (void)out_size; (void)ws_size;
}